// VoxelToElementBinary_26345329394107
// MI455X (gfx1250) — compile-verified
//
#include <hip/hip_runtime.h>

// CDNA5 / gfx1250 — wave32, 192MB L2, 23.3 TB/s HBM.
// Zero-FLOP scatter-fill: out[:] = EPS; out[idx[:]] = 1.0f + EPS.
// Memory-bound; optimized for L2 residency of the 128MB output and a
// non-temporal streamed read of the 64MB index array.

typedef int   v4i __attribute__((ext_vector_type(4)));
typedef float v4f __attribute__((ext_vector_type(4)));

#define EPS_F 1e-7f

// ---------------------------------------------------------------------------
// Kernel 1: fill output with EPS using 128-bit stores (default RT temporal
// hint: we WANT these lines resident in L2 — the scatter re-dirties ~100% of
// the output cachelines, and 128MB fits in the 192MB L2).
// ---------------------------------------------------------------------------
__global__ void VoxelToElementBinary_fill_eps(float* __restrict__ out,
                                              int n4, int n) {
  const int tid    = blockIdx.x * blockDim.x + threadIdx.x;
  const int stride = gridDim.x * blockDim.x;

  v4f v;
  v.x = EPS_F; v.y = EPS_F; v.z = EPS_F; v.w = EPS_F;

  v4f* __restrict__ out4 = (v4f*)out;
  for (int i = tid; i < n4; i += stride) {
    out4[i] = v;                       // global_store_b128, full-line writes
  }
  // Scalar tail (n not multiple of 4; here n = 33554432 so this is empty).
  for (int t = (n4 << 2) + tid; t < n; t += stride) {
    out[t] = EPS_F;
  }
}

// ---------------------------------------------------------------------------
// Kernel 2: scatter 1.0+EPS at each index.
//  - Indices are read once: non-temporal b128 loads (th:NT) so the 64MB index
//    stream does not evict the L2-resident output array.
//  - global_prefetch_b8 pulls the next grid-stride chunk of indices into L2
//    one iteration ahead (clamped in-bounds: WGP-scope prefetch is treated as
//    non-speculative per the ISA, so we never prefetch past the buffer).
//  - Scatter stores are idempotent (all lanes store the identical constant),
//    so plain global_store_b32 suffices — no atomics. ~All of them hit L2.
// ---------------------------------------------------------------------------
__global__ void VoxelToElementBinary_scatter(const int* __restrict__ idx,
                                             float* __restrict__ out,
                                             int n4, int n) {
  const float ONE = 1.0f + EPS_F;
  const int tid    = blockIdx.x * blockDim.x + threadIdx.x;
  const int stride = gridDim.x * blockDim.x;

  const v4i* __restrict__ idx4 = (const v4i*)idx;

  for (int i = tid; i < n4; i += stride) {
    // Prefetch next iteration's index quad into L2 (clamped in-bounds).
    int pf = i + stride;
    pf = (pf < n4) ? pf : i;
    __builtin_prefetch((const void*)(idx4 + pf), 0, 1);   // global_prefetch_b8

    v4i v = __builtin_nontemporal_load(idx4 + i);         // b128 load, th:NT

    out[v.x] = ONE;
    out[v.y] = ONE;
    out[v.z] = ONE;
    out[v.w] = ONE;
  }
  // Scalar tail for index count not a multiple of 4 (empty for 16777216).
  for (int t = (n4 << 2) + tid; t < n; t += stride) {
    int j = idx[t];
    out[j] = ONE;
  }
}

// ---------------------------------------------------------------------------
// Launch. Inputs (setup_inputs order): d_in[0]=voxel (f32, unused — output is
// value-independent of it), d_in[1]=cell_indices (i32), d_in[2]=num_elements
// (scalar; equal to out_size, which we use directly host-side).
// ---------------------------------------------------------------------------
extern "C" void kernel_launch(void* const* d_in, const int* in_sizes, int n_in,
                              void* d_out, int out_size, void* d_ws, size_t ws_size,
                              hipStream_t stream) {
  const int* idx = (const int*)d_in[1];
  float* out     = (float*)d_out;
  const int n_idx = in_sizes[1];      // 256^3 = 16,777,216
  const int n_out = out_size;         // 33,554,432

  const dim3 block(256);              // 8 wave32 waves per block

  // Fill: 8.4M float4 stores. 8192 blocks * 256 threads = 2M threads,
  // ~4 b128 stores each — enough in-flight stores to saturate HBM/L2.
  const int n4_out = n_out >> 2;
  VoxelToElementBinary_fill_eps<<<dim3(8192), block, 0, stream>>>(out, n4_out, n_out);

  // Scatter: 4.2M index quads. 4096 blocks * 256 threads = 1M threads,
  // ~4 iterations each so the prefetch-ahead has work to overlap.
  const int n4_idx = n_idx >> 2;
  VoxelToElementBinary_scatter<<<dim3(4096), block, 0, stream>>>(idx, out, n4_idx, n_idx);
}